// YoloXLoss_24481313587573
// MI455X (gfx1250) — compile-verified
//
#include <hip/hip_runtime.h>
#include <cfloat>
#include <cstdint>

#define NCLS   80
#define NCH    85
#define NANCH  8400
#define BATCH  32
#define NGT    16
#define KCAND  10
#define RADIUS 2.5f

typedef float v2f __attribute__((ext_vector_type(2)));
typedef float v8f __attribute__((ext_vector_type(8)));

// ---- workspace layout (float units) ----
static constexpr size_t OFF_PBOX  = 0;                                   // B*N*4
static constexpr size_t OFF_PCONF = OFF_PBOX  + (size_t)BATCH*NANCH*4;   // B*N
static constexpr size_t OFF_COST  = OFF_PCONF + (size_t)BATCH*NANCH;     // B*G*N
static constexpr size_t OFF_IOU   = OFF_COST  + (size_t)BATCH*NGT*NANCH; // B*G*N (raw iou)
static constexpr size_t OFF_MASK  = OFF_IOU   + (size_t)BATCH*NGT*NANCH; // B*N  (uint bitmask)
static constexpr size_t OFF_PART  = OFF_MASK  + (size_t)BATCH*NANCH;     // NPART*4
#define NBLK3_X 33
#define NPART   (BATCH * NBLK3_X)

__device__ __forceinline__ float bce_logit(float x, float t) {
  return fmaxf(x, 0.f) - x * t + log1pf(expf(-fabsf(x)));
}

__device__ __forceinline__ const float* anchor_base(int a, int b,
    const float* p8, const float* p16, const float* p32,
    int& idx, int& w, float& s, int& hw)
{
  const float* src;
  if (a < 6400)      { idx = a;        w = 80; s = 8.f;  src = p8;  }
  else if (a < 8000) { idx = a - 6400; w = 40; s = 16.f; src = p16; }
  else               { idx = a - 8000; w = 20; s = 32.f; src = p32; }
  hw = w * w;
  return src + (size_t)b * NCH * hw + idx;
}

// ---------------- init: zero mask + partials ----------------
__global__ __launch_bounds__(256) void k_init(float* ws) {
  int i = blockIdx.x * 256 + threadIdx.x;
  if (i < BATCH * NANCH) ((unsigned*)(ws + OFF_MASK))[i] = 0u;
  if (i < NPART * 4)     ws[OFF_PART + i] = 0.f;
}

// ---------------- cost kernel: decode + WMMA cls-cost + iou ----------------
// block = 256 threads = 8 waves; each wave handles a 16-anchor tile for one batch.
__global__ __launch_bounds__(256) void k_cost(const float* __restrict__ p8,
                                              const float* __restrict__ p16,
                                              const float* __restrict__ p32,
                                              const float* __restrict__ gtb,
                                              const int*   __restrict__ gtc,
                                              float* __restrict__ ws)
{
  __shared__ float s_gt[NGT * 4];
  __shared__ int   s_gc[NGT];
  const int b   = blockIdx.y;
  const int tid = threadIdx.x;
  if (tid < NGT * 4) s_gt[tid] = gtb[b * NGT * 4 + tid];
  if (tid < NGT)     s_gc[tid] = gtc[b * NGT + tid];
  __syncthreads();

  const int wave = tid >> 5;
  const int l    = tid & 31;
  const int col  = l & 15;    // A-row (gt) & B-col (anchor) index for this lane
  const int hi   = l >> 4;    // lane half
  const int tile = blockIdx.x * 8 + wave;
  const int a_real = tile * 16 + col;
  const bool oob = a_real >= NANCH;
  const int a = oob ? (NANCH - 1) : a_real;

  int idx, w, hw; float s;
  const float* base = anchor_base(a, b, p8, p16, p32, idx, w, s, hw);
  const int gy = idx / w, gx = idx - gy * w;

  const float tx = base[(size_t)0 * hw], ty = base[(size_t)1 * hw];
  const float tw = base[(size_t)2 * hw], th = base[(size_t)3 * hw];
  const float tconf = base[(size_t)4 * hw];
  const float bx = (tx + (float)gx) * s, by = (ty + (float)gy) * s;
  const float bw = expf(tw) * s,         bh = expf(th) * s;
  const float sig_conf = 1.f / (1.f + expf(-tconf));

  if (!oob && hi == 0) {
    float* pb = ws + OFF_PBOX + ((size_t)b * NANCH + a) * 4;
    pb[0] = bx; pb[1] = by; pb[2] = bw; pb[3] = bh;
    ws[OFF_PCONF + (size_t)b * NANCH + a] = tconf;
  }

  const int my_gt = s_gc[col];  // gt class for A-matrix row = l&15

  // WMMA chains:  acc1 = onehot x (logp-log1mp),  acc2 = ones x log1mp
  v8f acc1 = {0.f,0.f,0.f,0.f,0.f,0.f,0.f,0.f};
  v8f acc2 = {0.f,0.f,0.f,0.f,0.f,0.f,0.f,0.f};
  for (int k0 = 0; k0 < NCLS; k0 += 4) {
    const int c0 = k0 + (hi << 1);   // this lane's K indices: c0, c0+1
    const float z0 = base[(size_t)(5 + c0) * hw];
    const float z1 = base[(size_t)(6 + c0) * hw];
    const float p0 = sqrtf((1.f / (1.f + expf(-z0))) * sig_conf);
    const float p1 = sqrtf((1.f / (1.f + expf(-z1))) * sig_conf);
    const float lp0 = fmaxf(logf(fmaxf(p0,       1e-12f)), -100.f);
    const float lq0 = fmaxf(logf(fmaxf(1.f - p0, 1e-12f)), -100.f);
    const float lp1 = fmaxf(logf(fmaxf(p1,       1e-12f)), -100.f);
    const float lq1 = fmaxf(logf(fmaxf(1.f - p1, 1e-12f)), -100.f);
    v2f bA; bA.x = lp0 - lq0; bA.y = lp1 - lq1;
    v2f bS; bS.x = lq0;       bS.y = lq1;
    v2f aO; aO.x = (my_gt == c0)     ? 1.f : 0.f;
            aO.y = (my_gt == c0 + 1) ? 1.f : 0.f;
    v2f aI; aI.x = 1.f; aI.y = 1.f;
    acc1 = __builtin_amdgcn_wmma_f32_16x16x4_f32(false, aO, false, bA, 0, acc1, false, false);
    acc2 = __builtin_amdgcn_wmma_f32_16x16x4_f32(false, aI, false, bS, 0, acc2, false, false);
  }
  float A1[8], A2[8];
#pragma unroll
  for (int r = 0; r < 8; ++r) { A1[r] = acc1[r]; A2[r] = acc2[r]; }

  // geometric masks for all 16 gts (per-anchor union needs all of them)
  const float cx = ((float)gx + 0.5f) * s, cy = ((float)gy + 0.5f) * s;
  const float rs = RADIUS * s;
  unsigned inter_mask = 0; bool uni = false;
#pragma unroll
  for (int g = 0; g < NGT; ++g) {
    const float gx0 = s_gt[g*4+0], gy0 = s_gt[g*4+1];
    const float gw0 = s_gt[g*4+2], gh0 = s_gt[g*4+3];
    const float hx = gw0 * 0.5f, hy = gh0 * 0.5f;
    const float d1 = fminf(fminf(cx - (gx0 - hx), cy - (gy0 - hy)),
                           fminf((gx0 + hx) - cx, (gy0 + hy) - cy));
    const float d2 = fminf(fminf(cx - (gx0 - rs), cy - (gy0 - rs)),
                           fminf((gx0 + rs) - cx, (gy0 + rs) - cy));
    const bool ib = d1 > 0.f, im = d2 > 0.f;
    uni = uni || ib || im;
    if (ib && im) inter_mask |= (1u << g);
  }

  const float areaP = bw * bh;
  const float pxm = bx - bw * 0.5f, pym = by - bh * 0.5f;
  const float pxM = bx + bw * 0.5f, pyM = by + bh * 0.5f;
  float* costW = ws + OFF_COST + (size_t)b * NGT * NANCH;
  float* iouW  = ws + OFF_IOU  + (size_t)b * NGT * NANCH;
#pragma unroll
  for (int r = 0; r < 8; ++r) {
    const int g = r + hi * 8;   // D layout: vgpr r, lane half -> row
    const float gx0 = s_gt[g*4+0], gy0 = s_gt[g*4+1];
    const float gw0 = s_gt[g*4+2], gh0 = s_gt[g*4+3];
    const float tlx = fmaxf(gx0 - gw0*0.5f, pxm), tly = fmaxf(gy0 - gh0*0.5f, pym);
    const float brx = fminf(gx0 + gw0*0.5f, pxM), bry = fminf(gy0 + gh0*0.5f, pyM);
    const float iw = fmaxf(brx - tlx, 0.f), ih = fmaxf(bry - tly, 0.f);
    const float inter = iw * ih;
    const float iou = inter / (gw0 * gh0 + areaP - inter + 1e-16f);
    const float cls_cost = -(A1[r] + A2[r]);
    float c = cls_cost + 3.f * (-logf(iou + 1e-8f))
            + (((inter_mask >> g) & 1u) ? 0.f : 100000.f);
    c = uni ? c : 1e15f;   // 1e15 sentinel encodes !union
    if (!oob) {
      costW[(size_t)g * NANCH + a] = c;
      iouW [(size_t)g * NANCH + a] = iou;   // raw iou (ref uses raw for match_iou)
    }
  }
}

// ---------------- top-k per (b,g): dyn_k + candidate scatter ----------------
__global__ __launch_bounds__(256) void k_topk(float* __restrict__ ws)
{
  const int g = blockIdx.x, b = blockIdx.y, t = threadIdx.x;
  const float* costR = ws + OFF_COST + ((size_t)b * NGT + g) * NANCH;
  const float* iouR  = ws + OFF_IOU  + ((size_t)b * NGT + g) * NANCH;

  float vi[KCAND]; float cv[KCAND]; int ci[KCAND];
#pragma unroll
  for (int k = 0; k < KCAND; ++k) { vi[k] = -1.f; cv[k] = FLT_MAX; ci[k] = 0x7fffffff; }

  for (int a = t; a < NANCH; a += 256) {
    const float cst = costR[a];
    float vio = (cst == 1e15f) ? 0.f : iouR[a];   // masked iou for dyn_k
#pragma unroll
    for (int k = 0; k < KCAND; ++k)
      if (vio > vi[k]) { float tm = vi[k]; vi[k] = vio; vio = tm; }
    float vc = cst; int id = a;
#pragma unroll
    for (int k = 0; k < KCAND; ++k) {
      const bool lt = (vc < cv[k]) || (vc == cv[k] && id < ci[k]);
      if (lt) { float tf = cv[k]; cv[k] = vc; vc = tf;
                int   ti = ci[k]; ci[k] = id; id = ti; }
    }
  }

  __shared__ float s_vi[256 * KCAND];
  __shared__ float s_cv[256 * KCAND];
  __shared__ int   s_ci[256 * KCAND];
#pragma unroll
  for (int k = 0; k < KCAND; ++k) {
    s_vi[t*KCAND+k] = vi[k]; s_cv[t*KCAND+k] = cv[k]; s_ci[t*KCAND+k] = ci[k];
  }
  __syncthreads();

  for (int str = 128; str >= 1; str >>= 1) {
    if (t < str) {
      int i = 0, j = 0; float ov[KCAND];
#pragma unroll
      for (int k = 0; k < KCAND; ++k) {
        const float av = s_vi[t*KCAND + i], bv = s_vi[(t+str)*KCAND + j];
        if (av >= bv) { ov[k] = av; ++i; } else { ov[k] = bv; ++j; }
      }
      int i2 = 0, j2 = 0; float oc[KCAND]; int oi[KCAND];
#pragma unroll
      for (int k = 0; k < KCAND; ++k) {
        const float ac = s_cv[t*KCAND + i2]; const int ai = s_ci[t*KCAND + i2];
        const float bc = s_cv[(t+str)*KCAND + j2]; const int bi = s_ci[(t+str)*KCAND + j2];
        const bool takeA = (ac < bc) || (ac == bc && ai < bi);
        if (takeA) { oc[k] = ac; oi[k] = ai; ++i2; } else { oc[k] = bc; oi[k] = bi; ++j2; }
      }
#pragma unroll
      for (int k = 0; k < KCAND; ++k) {
        s_vi[t*KCAND+k] = ov[k]; s_cv[t*KCAND+k] = oc[k]; s_ci[t*KCAND+k] = oi[k];
      }
    }
    __syncthreads();
  }

  if (t == 0) {
    float ssum = 0.f;
#pragma unroll
    for (int k = 0; k < KCAND; ++k) ssum += s_vi[k];
    int dynk = (int)ssum;                 // trunc toward zero, matches astype(int32)
    if (dynk < 1) dynk = 1; if (dynk > KCAND) dynk = KCAND;
    unsigned* mask = (unsigned*)(ws + OFF_MASK) + (size_t)b * NANCH;
    for (int k = 0; k < dynk; ++k) atomicOr(&mask[s_ci[k]], 1u << g);
  }
}

// ---------------- assignment resolution + loss partials ----------------
__global__ __launch_bounds__(256) void k_loss(const float* __restrict__ p8,
                                              const float* __restrict__ p16,
                                              const float* __restrict__ p32,
                                              const float* __restrict__ gtb,
                                              const int*   __restrict__ gtc,
                                              float* __restrict__ ws)
{
  const int b = blockIdx.y;
  const int a = blockIdx.x * 256 + threadIdx.x;
  float lb = 0.f, lcf = 0.f, lcl = 0.f, nfg = 0.f;

  if (a < NANCH) {
    const unsigned m = ((const unsigned*)(ws + OFF_MASK))[(size_t)b * NANCH + a];
    const int cnt = __popc(m);
    const float* costC = ws + OFF_COST + (size_t)b * NGT * NANCH;
    int gsel = 0; const bool fg = cnt > 0;
    if (cnt == 1) gsel = __ffs(m) - 1;
    else if (cnt > 1) {                       // ref: argmin over ALL gts, first min wins
      float best = costC[a];
      for (int g = 1; g < NGT; ++g) {
        const float c = costC[(size_t)g * NANCH + a];
        if (c < best) { best = c; gsel = g; }
      }
    }
    const float pconf = ws[OFF_PCONF + (size_t)b * NANCH + a];
    const float fgf = fg ? 1.f : 0.f;
    lcf = bce_logit(pconf, fgf);
    if (fg) {
      nfg = 1.f;
      const float miou = ws[OFF_IOU + ((size_t)b * NGT + gsel) * NANCH + a];
      const float* pb = ws + OFF_PBOX + ((size_t)b * NANCH + a) * 4;
      const float bx = pb[0], by = pb[1], bw = pb[2], bh = pb[3];
      const float gx0 = gtb[(b*NGT+gsel)*4+0], gy0 = gtb[(b*NGT+gsel)*4+1];
      const float gw0 = gtb[(b*NGT+gsel)*4+2], gh0 = gtb[(b*NGT+gsel)*4+3];
      const float tlx = fmaxf(bx - bw*0.5f, gx0 - gw0*0.5f);
      const float tly = fmaxf(by - bh*0.5f, gy0 - gh0*0.5f);
      const float brx = fminf(bx + bw*0.5f, gx0 + gw0*0.5f);
      const float bry = fminf(by + bh*0.5f, gy0 + gh0*0.5f);
      const float inter = fmaxf(brx - tlx, 0.f) * fmaxf(bry - tly, 0.f);
      const float iou_e = inter / (bw*bh + gw0*gh0 - inter + 1e-16f);
      lb = 1.f - iou_e * iou_e;
      const int tc = gtc[b * NGT + gsel];
      int idx, w, hw; float s;
      const float* base = anchor_base(a, b, p8, p16, p32, idx, w, s, hw);
      for (int c = 0; c < NCLS; ++c) {
        const float x = base[(size_t)(5 + c) * hw];
        lcl += bce_logit(x, (c == tc) ? miou : 0.f);
      }
    }
  }

  // deterministic fixed-tree block reduction of 4 values
  __shared__ float red[256];
  float vals[4] = { lb, lcf, lcl, nfg };
  float* part = ws + OFF_PART + ((size_t)(b * NBLK3_X + blockIdx.x)) * 4;
#pragma unroll
  for (int q = 0; q < 4; ++q) {
    red[threadIdx.x] = vals[q];
    __syncthreads();
    for (int st = 128; st >= 1; st >>= 1) {
      if (threadIdx.x < st) red[threadIdx.x] += red[threadIdx.x + st];
      __syncthreads();
    }
    if (threadIdx.x == 0) part[q] = red[0];
    __syncthreads();
  }
}

// ---------------- final deterministic reduction ----------------
__global__ __launch_bounds__(256) void k_final(const float* __restrict__ ws,
                                               float* __restrict__ out)
{
  const int t = threadIdx.x;
  float s0 = 0.f, s1 = 0.f, s2 = 0.f, s3 = 0.f;
  for (int i = t; i < NPART; i += 256) {
    const float* p = ws + OFF_PART + (size_t)i * 4;
    s0 += p[0]; s1 += p[1]; s2 += p[2]; s3 += p[3];
  }
  __shared__ float r0[256], r1[256], r2[256], r3[256];
  r0[t] = s0; r1[t] = s1; r2[t] = s2; r3[t] = s3;
  __syncthreads();
  for (int st = 128; st >= 1; st >>= 1) {
    if (t < st) { r0[t] += r0[t+st]; r1[t] += r1[t+st]; r2[t] += r2[t+st]; r3[t] += r3[t+st]; }
    __syncthreads();
  }
  if (t == 0) out[0] = (5.f * r0[0] + r1[0] + r2[0]) / fmaxf(r3[0], 1.f);
}

extern "C" void kernel_launch(void* const* d_in, const int* in_sizes, int n_in,
                              void* d_out, int out_size, void* d_ws, size_t ws_size,
                              hipStream_t stream) {
  const float* p8  = (const float*)d_in[0];
  const float* p16 = (const float*)d_in[1];
  const float* p32 = (const float*)d_in[2];
  const float* gtb = (const float*)d_in[3];
  const int*   gtc = (const int*)  d_in[4];
  float* ws  = (float*)d_ws;
  float* out = (float*)d_out;

  k_init <<<dim3((BATCH*NANCH + 255)/256), dim3(256), 0, stream>>>(ws);
  k_cost <<<dim3(66,      BATCH), dim3(256), 0, stream>>>(p8, p16, p32, gtb, gtc, ws);
  k_topk <<<dim3(NGT,     BATCH), dim3(256), 0, stream>>>(ws);
  k_loss <<<dim3(NBLK3_X, BATCH), dim3(256), 0, stream>>>(p8, p16, p32, gtb, gtc, ws);
  k_final<<<dim3(1), dim3(256), 0, stream>>>(ws, out);
}